// QuantumLayer_57269093924909
// MI455X (gfx1250) — compile-verified
//
#include <hip/hip_runtime.h>
#include <cstdint>

#define N_QUBITS   25
#define N_LAYERS   10
#define TOTAL      50000000u            // 2,000,000 * 25 f32 elements = 2^7 * 5^8
#define BLOCK      128                  // 4 waves (wave32)
#define GRID       3125
#define STRIDE     (GRID * BLOCK)       // 400,000  (divisible by 25)
#define UNROLL     5
#define CHUNK      (UNROLL * STRIDE)    // 2,000,000 ; TOTAL / CHUNK = 25 exact
#define N_ITERS    (TOTAL / CHUNK)      // 25
#define N_PARAMS   (N_QUBITS * N_LAYERS * 3) // 750
#define INV2PI     0.15915494309189535f

// Hardware tanh (gfx1250 V_TANH_F32) with a 2-trans fallback.
#if __has_builtin(__builtin_amdgcn_tanhf)
__device__ __forceinline__ float fast_tanh(float t) { return __builtin_amdgcn_tanhf(t); }
#elif __has_builtin(__builtin_amdgcn_tanh_f32)
__device__ __forceinline__ float fast_tanh(float t) { return __builtin_amdgcn_tanh_f32(t); }
#else
__device__ __forceinline__ float fast_tanh(float t) {
    // tanh(t) = 1 - 2/(exp(2t)+1), exp(2t) = exp2(2*log2(e)*t)
    float p = __builtin_amdgcn_exp2f(2.8853900817779268f * t);
    return 1.0f - 2.0f * __builtin_amdgcn_rcpf(p + 1.0f);
}
#endif

// Full column-update chain; all params in registers.
__device__ __forceinline__ float chain(float xv, float sc,
                                       const float* __restrict__ ry,
                                       const float* __restrict__ rz) {
    float e = fast_tanh(xv * sc);
#pragma unroll
    for (int l = 0; l < N_LAYERS; ++l) {
        // ry/rz pre-scaled by 1/(2pi): v_cos_f32/v_sin_f32 evaluate trig(2*pi*src)
        e = __builtin_amdgcn_cosf(e * ry[l]) * __builtin_amdgcn_sinf(e * rz[l]);
    }
    return e;
}

__global__ __launch_bounds__(BLOCK) void quantum_layer_kernel(
    const float* __restrict__ x,
    const float* __restrict__ qp,
    float* __restrict__ out)
{
    __shared__ float s_raw[N_PARAMS];            // raw params via CDNA5 async copy
    __shared__ float s_scale[N_QUBITS];          // qp[0:25] (radians, tanh arg)
    __shared__ float s_ry[N_LAYERS][N_QUBITS];   // ry * 1/(2pi)
    __shared__ float s_rz[N_LAYERS][N_QUBITS];   // rz * 1/(2pi)

    const int tid = threadIdx.x;

    // ---- CDNA5 async global->LDS copy of the parameter block (ASYNCcnt path) ----
    for (int i = tid; i < N_PARAMS; i += BLOCK) {
        uint32_t lds_off = (uint32_t)(uintptr_t)(&s_raw[i]); // low 32b of generic LDS addr = LDS offset
        const float* g = qp + i;
        asm volatile("global_load_async_to_lds_b32 %0, %1, off"
                     :: "v"(lds_off), "v"(g) : "memory");
    }
    asm volatile("s_wait_asynccnt 0" ::: "memory");
    __syncthreads();

    // ---- rescale pass: fold 1/(2pi) into rotation params ----
    for (int i = tid; i < N_QUBITS + N_LAYERS * 2 * N_QUBITS; i += BLOCK) {
        if (i < N_QUBITS) {
            s_scale[i] = s_raw[i];
        } else {
            int j = i - N_QUBITS;                 // 0..499
            int layer = j / (2 * N_QUBITS);
            int k = j - layer * (2 * N_QUBITS);   // 0..49: [0,25)=ry, [25,50)=rz
            float v = s_raw[layer * (3 * N_QUBITS) + k] * INV2PI;
            if (k < N_QUBITS) s_ry[layer][k] = v;
            else              s_rz[layer][k - N_QUBITS] = v;
        }
    }
    __syncthreads();

    // STRIDE % 25 == 0 => column invariant for this thread across the whole loop.
    const uint32_t idx0 = (uint32_t)blockIdx.x * BLOCK + (uint32_t)tid;
    const uint32_t c    = idx0 % (uint32_t)N_QUBITS;

    const float sc = s_scale[c];
    float ry[N_LAYERS], rz[N_LAYERS];
#pragma unroll
    for (int l = 0; l < N_LAYERS; ++l) { ry[l] = s_ry[l][c]; rz[l] = s_rz[l][c]; }

    // Exact tiling: TOTAL == N_ITERS * UNROLL * STRIDE, so NO bounds checks.
    // Each of the 5 per-iteration elements is a fully coalesced b32 access
    // (wave32 x 4B = 128B/instr); 5 independent trig chains give trans-pipe ILP.
    uint32_t i = idx0;
#pragma unroll 1
    for (uint32_t it = 0; it < N_ITERS; ++it, i += CHUNK) {
        float xv[UNROLL], ev[UNROLL];
#pragma unroll
        for (int j = 0; j < UNROLL; ++j)
            xv[j] = __builtin_nontemporal_load(x + i + (uint32_t)j * STRIDE);
#pragma unroll
        for (int j = 0; j < UNROLL; ++j)
            ev[j] = chain(xv[j], sc, ry, rz);
#pragma unroll
        for (int j = 0; j < UNROLL; ++j)
            __builtin_nontemporal_store(ev[j], out + i + (uint32_t)j * STRIDE);
    }
}

extern "C" void kernel_launch(void* const* d_in, const int* in_sizes, int n_in,
                              void* d_out, int out_size, void* d_ws, size_t ws_size,
                              hipStream_t stream) {
    (void)in_sizes; (void)n_in; (void)d_ws; (void)ws_size; (void)out_size;
    const float* x  = (const float*)d_in[0];   // [2e6, 25] f32
    const float* qp = (const float*)d_in[1];   // [750] f32
    float* out = (float*)d_out;                // [2e6, 25] f32

    quantum_layer_kernel<<<GRID, BLOCK, 0, stream>>>(x, qp, out);
}